// EvaGNN_16260746182785
// MI455X (gfx1250) — compile-verified
//
#include <hip/hip_runtime.h>
#include <hip/hip_bf16.h>
#include <math.h>

// CDNA5 f32 WMMA fragment types: 16x16x4_f32 -> A/B = 2 VGPRs, C/D = 8 VGPRs (wave32)
typedef __attribute__((ext_vector_type(2))) float v2f;
typedef __attribute__((ext_vector_type(8))) float v8f;

#define IN_DIM 128
#define HID    16

// ---------------- degree / norm ----------------

__global__ void k_init_deg(float* __restrict__ deg, int n) {
  int i = blockIdx.x * blockDim.x + threadIdx.x;
  if (i < n) deg[i] = 1.0f;  // self-loop contribution
}

__global__ void k_deg_edges(const int* __restrict__ dst, float* __restrict__ deg, int nE) {
  int e = blockIdx.x * blockDim.x + threadIdx.x;
  if (e < nE) atomicAdd(&deg[dst[e]], 1.0f);
}

__global__ void k_dis(const float* __restrict__ deg, float* __restrict__ dis, int n) {
  int i = blockIdx.x * blockDim.x + threadIdx.x;
  if (i < n) dis[i] = rsqrtf(deg[i]);  // deg >= 1 always (self-loops)
}

// ---------------- GEMM1: h[N,16] = x[N,128] @ W1[128,16] via V_WMMA_F32_16X16X4_F32 ----------------
// One wave per 16-row tile; 32 WMMA steps of K=4.
// A layout (16x4 f32): lane L (L<16): M=L, K={0,1} in v0/v1; lane L+16: K={2,3}.
// B layout (4x16 f32): v0 = rows K=0 (lanes 0-15) / K=2 (lanes 16-31); v1 = K=1 / K=3.
// C/D (16x16 f32): vgpr r, lanes 0-15 -> M=r, lanes 16-31 -> M=r+8; N=lane%16.

__global__ void k_gemm1_wmma(const float* __restrict__ x, const float* __restrict__ W1,
                             float* __restrict__ h, int n) {
  const int lane = threadIdx.x & 31;
  const int wave = threadIdx.x >> 5;
  const int tile = blockIdx.x * (blockDim.x >> 5) + wave;
  const int row0 = tile << 4;
  if (row0 >= n) return;               // wave-uniform guard: EXEC all-1s at WMMA
  const int m  = lane & 15;            // A row / B col / D col
  const int kh = lane >> 4;            // K half: +0 or +2 within each K=4 group
  const float* xrow = x + (size_t)(row0 + m) * IN_DIM + 2 * kh;

  v8f c = {0.f, 0.f, 0.f, 0.f, 0.f, 0.f, 0.f, 0.f};
#pragma unroll
  for (int k0 = 0; k0 < IN_DIM / 4; ++k0) {
    const int kk = 4 * k0 + 2 * kh;
    const float2 av = *reinterpret_cast<const float2*>(xrow + 4 * k0);
    v2f a; a.x = av.x; a.y = av.y;
    v2f b; b.x = W1[(kk + 0) * HID + m];
           b.y = W1[(kk + 1) * HID + m];
    // 8 args: (neg_a, A, neg_b, B, c_mod, C, reuse_a, reuse_b)
    c = __builtin_amdgcn_wmma_f32_16x16x4_f32(false, a, false, b, (short)0, c, false, false);
  }
  float* hout = h + (size_t)(row0 + 8 * kh) * HID + m;
#pragma unroll
  for (int r = 0; r < 8; ++r)
    hout[(size_t)r * HID] = c[r];
}

// ---------------- GEMM2: h2[N,2] = h1[N,16] @ W2[16,2], N-dim zero-padded to 16 ----------------

__global__ void k_gemm2_wmma(const float* __restrict__ h1, const float* __restrict__ W2,
                             float* __restrict__ h2, int n) {
  const int lane = threadIdx.x & 31;
  const int wave = threadIdx.x >> 5;
  const int tile = blockIdx.x * (blockDim.x >> 5) + wave;
  const int row0 = tile << 4;
  if (row0 >= n) return;               // wave-uniform
  const int m  = lane & 15;
  const int kh = lane >> 4;
  const float bmask = (m < 2) ? 1.0f : 0.0f;  // pad B cols 2..15 with zeros
  const int   col   = (m < 2) ? m : 0;
  const float* arow = h1 + (size_t)(row0 + m) * HID + 2 * kh;

  v8f c = {0.f, 0.f, 0.f, 0.f, 0.f, 0.f, 0.f, 0.f};
#pragma unroll
  for (int k0 = 0; k0 < HID / 4; ++k0) {
    const int kk = 4 * k0 + 2 * kh;
    const float2 av = *reinterpret_cast<const float2*>(arow + 4 * k0);
    v2f a; a.x = av.x; a.y = av.y;
    v2f b; b.x = W2[(kk + 0) * 2 + col] * bmask;
           b.y = W2[(kk + 1) * 2 + col] * bmask;
    c = __builtin_amdgcn_wmma_f32_16x16x4_f32(false, a, false, b, (short)0, c, false, false);
  }
  if (m < 2) {                          // divergence after WMMA is fine
#pragma unroll
    for (int r = 0; r < 8; ++r)
      h2[(size_t)(row0 + r + 8 * kh) * 2 + m] = c[r];
  }
}

// ---------------- aggregation ----------------

// Seed accumulator with the self-loop message: out1[i] = h[i] * dis[i]^2
__global__ void k_self1(const float* __restrict__ h, const float* __restrict__ dis,
                        float* __restrict__ out1, int n) {
  int i = blockIdx.x * blockDim.x + threadIdx.x;
  if (i >= n) return;
  float s = dis[i]; s = s * s;
  const float4* hp = reinterpret_cast<const float4*>(h + (size_t)i * HID);
  float4* op = reinterpret_cast<float4*>(out1 + (size_t)i * HID);
#pragma unroll
  for (int q = 0; q < 4; ++q) {
    float4 v = hp[q];
    v.x *= s; v.y *= s; v.z *= s; v.w *= s;
    op[q] = v;
  }
}

// 4 threads per edge, each handles 4 of the 16 features (float4 gather + 4 f32 atomics)
__global__ void k_scatter1(const int* __restrict__ src, const int* __restrict__ dst,
                           const float* __restrict__ dis, const float* __restrict__ h,
                           float* __restrict__ out1, int nE) {
  long long t = (long long)blockIdx.x * blockDim.x + threadIdx.x;
  int e = (int)(t >> 2);
  if (e >= nE) return;
  int q = (int)(t & 3);
  int s = src[e], d = dst[e];
  float norm = dis[s] * dis[d];
  float4 v = *reinterpret_cast<const float4*>(h + (size_t)s * HID + 4 * q);
  float* o = out1 + (size_t)d * HID + 4 * q;
  atomicAdd(o + 0, v.x * norm);
  atomicAdd(o + 1, v.y * norm);
  atomicAdd(o + 2, v.z * norm);
  atomicAdd(o + 3, v.w * norm);
}

__global__ void k_relu_bias(float* __restrict__ out1, const float* __restrict__ b1, int n) {
  long long t = (long long)blockIdx.x * blockDim.x + threadIdx.x;  // n*16 threads
  if (t >= (long long)n * HID) return;
  int j = (int)(t & (HID - 1));
  float v = out1[t] + b1[j];
  out1[t] = v > 0.f ? v : 0.f;
}

__global__ void k_self2(const float* __restrict__ h2, const float* __restrict__ dis,
                        float* __restrict__ out2, int n) {
  int i = blockIdx.x * blockDim.x + threadIdx.x;
  if (i >= n) return;
  float s = dis[i]; s = s * s;
  out2[(size_t)i * 2 + 0] = h2[(size_t)i * 2 + 0] * s;
  out2[(size_t)i * 2 + 1] = h2[(size_t)i * 2 + 1] * s;
}

__global__ void k_scatter2(const int* __restrict__ src, const int* __restrict__ dst,
                           const float* __restrict__ dis, const float* __restrict__ h2,
                           float* __restrict__ out2, int nE) {
  int e = blockIdx.x * blockDim.x + threadIdx.x;
  if (e >= nE) return;
  int s = src[e], d = dst[e];
  float norm = dis[s] * dis[d];
  atomicAdd(&out2[(size_t)d * 2 + 0], h2[(size_t)s * 2 + 0] * norm);
  atomicAdd(&out2[(size_t)d * 2 + 1], h2[(size_t)s * 2 + 1] * norm);
}

__global__ void k_logsoftmax(const float* __restrict__ out2, const float* __restrict__ b2,
                             float* __restrict__ y, int n) {
  int i = blockIdx.x * blockDim.x + threadIdx.x;
  if (i >= n) return;
  float a0 = out2[(size_t)i * 2 + 0] + b2[0];
  float a1 = out2[(size_t)i * 2 + 1] + b2[1];
  float mx = fmaxf(a0, a1);
  float lse = mx + logf(expf(a0 - mx) + expf(a1 - mx));
  y[(size_t)i * 2 + 0] = a0 - lse;
  y[(size_t)i * 2 + 1] = a1 - lse;
}

// ---------------- launch ----------------

extern "C" void kernel_launch(void* const* d_in, const int* in_sizes, int n_in,
                              void* d_out, int out_size, void* d_ws, size_t ws_size,
                              hipStream_t stream) {
  const float* x  = (const float*)d_in[0];
  const int*   ei = (const int*)d_in[1];     // [2, E] row-major int32
  const float* W1 = (const float*)d_in[2];   // [128,16]
  const float* b1 = (const float*)d_in[3];   // [16]
  const float* W2 = (const float*)d_in[4];   // [16,2]
  const float* b2 = (const float*)d_in[5];   // [2]

  const int n  = in_sizes[0] / IN_DIM;       // 200000 (divisible by 16)
  const int nE = in_sizes[1] / 2;            // 6.4M
  const int* src = ei;
  const int* dst = ei + nE;

  // workspace partition (floats): deg[n] dis[n] h[n*16] out1[n*16] h2[n*2] out2[n*2]
  float* ws   = (float*)d_ws;
  float* deg  = ws;
  float* dis  = deg + n;
  float* h    = dis + n;
  float* out1 = h    + (size_t)n * HID;
  float* h2   = out1 + (size_t)n * HID;
  float* out2 = h2   + (size_t)n * 2;
  float* y    = (float*)d_out;

  const int TB = 256;                        // 8 wave32 waves per block
  dim3 blk(TB);
  int gN   = (n + TB - 1) / TB;
  int gE   = (nE + TB - 1) / TB;
  int gE4  = (int)(((long long)nE * 4 + TB - 1) / TB);
  int gN16 = (int)(((long long)n * HID + TB - 1) / TB);
  int tiles = (n + 15) / 16;
  int gT   = (tiles + (TB / 32) - 1) / (TB / 32);

  k_init_deg  <<<gN,  blk, 0, stream>>>(deg, n);
  k_deg_edges <<<gE,  blk, 0, stream>>>(dst, deg, nE);
  k_dis       <<<gN,  blk, 0, stream>>>(deg, dis, n);
  k_gemm1_wmma<<<gT,  blk, 0, stream>>>(x, W1, h, n);
  k_self1     <<<gN,  blk, 0, stream>>>(h, dis, out1, n);
  k_scatter1  <<<gE4, blk, 0, stream>>>(src, dst, dis, h, out1, nE);
  k_relu_bias <<<gN16,blk, 0, stream>>>(out1, b1, n);
  k_gemm2_wmma<<<gT,  blk, 0, stream>>>(out1, W2, h2, n);
  k_self2     <<<gN,  blk, 0, stream>>>(h2, dis, out2, n);
  k_scatter2  <<<gE,  blk, 0, stream>>>(src, dst, dis, h2, out2, nE);
  k_logsoftmax<<<gN,  blk, 0, stream>>>(out2, b2, y, n);
}